// Module2QueryNetwork_37520834298057
// MI455X (gfx1250) — compile-verified
//
#include <hip/hip_runtime.h>
#include <math.h>

#define EPS 1e-8f
#define NQ 16384
#define NB 128
#define NF 64
#define HD 128

typedef __attribute__((ext_vector_type(2))) float v2f;
typedef __attribute__((ext_vector_type(8))) float v8f;

// ---------------------------------------------------------------------------
// Prep kernel: probe normalization + rotation, eff_w = -softplus(w_raw),
// and a transpose of q_magnitude_weights. Everything written [f][b] so the
// main kernel stages LDS with flat coalesced float4 copies.
// 128 blocks x 64 threads.
// ---------------------------------------------------------------------------
__global__ void prep_kernel(const float* __restrict__ probes,
                            const float* __restrict__ angles,
                            const float* __restrict__ w_raw,
                            const float* __restrict__ mw,
                            float* __restrict__ wsPr,   // [NF][NB]
                            float* __restrict__ wsPi,   // [NF][NB]
                            float* __restrict__ wsW,    // [NF][NB]
                            float* __restrict__ wsMW) { // [NF][NB]
  __shared__ float red[NF];
  const int b = blockIdx.x;
  const int f = threadIdx.x; // 0..63

  const float x0 = probes[b * HD + f];       // real half
  const float x1 = probes[b * HD + NF + f];  // imag half
  red[f] = x0 * x0 + x1 * x1;
  __syncthreads();
  for (int off = NF / 2; off > 0; off >>= 1) {
    if (f < off) red[f] += red[f + off];
    __syncthreads();
  }
  const float inv = 1.0f / (sqrtf(red[0]) + EPS);
  const float pr = x0 * inv;
  const float pi = x1 * inv;
  const float ca = cosf(angles[f]);
  const float sa = sinf(angles[f]);
  wsPr[f * NB + b] = pr * ca - pi * sa;
  wsPi[f * NB + b] = pr * sa + pi * ca;

  const float x = w_raw[b * NF + f];
  wsW[f * NB + b] = -(fmaxf(x, 0.0f) + log1pf(expf(-fabsf(x)))); // -softplus
  wsMW[f * NB + b] = mw[b * NF + f];
}

// ---------------------------------------------------------------------------
// Main kernel: 256 threads (8 waves) per block. Block covers a 16q x 128b
// output stripe; wave w owns the 16-wide b-tile [16w, 16w+16).
// Q is staged + normalized ONCE per block (8x less global Q traffic than a
// per-tile scheme). Accumulator uses the WMMA 16x16 f32 C/D lane layout so
// the magnitude GEMM (v_wmma_f32_16x16x4_f32) and the scalar sqrt-distance
// loop share the same registers.
// ---------------------------------------------------------------------------
__global__ void __launch_bounds__(256)
qnet_kernel(const float* __restrict__ Q,
            const float* __restrict__ bias,
            const float* __restrict__ wsPr,
            const float* __restrict__ wsPi,
            const float* __restrict__ wsW,
            const float* __restrict__ wsMW,
            float* __restrict__ out) {
  __shared__ __align__(16) float  sQ[16][HD];     // raw Q tile
  __shared__ float  sInv[16];
  __shared__ __align__(16) float2 sQri[NF][16];   // interleaved (qr,qi) [f][m]
  __shared__ __align__(16) float  sQm[NF][16];    // |q| [f][m]
  __shared__ __align__(16) float  sPr[NF][NB];
  __shared__ __align__(16) float  sPi[NF][NB];
  __shared__ __align__(16) float  sW [NF][NB];
  __shared__ __align__(16) float  sMW[NF][NB];
  __shared__ float sB[NB];

  const int tid  = threadIdx.x;   // 0..255
  const int lane = tid & 31;
  const int wave = tid >> 5;      // 0..7
  const int q0 = blockIdx.x * 16;
  const int b0 = wave * 16;

  // ---- stage Q tile (16 x 128 floats) ----
  {
    const float4* g = (const float4*)(Q + (size_t)q0 * HD);
    float4* s = (float4*)&sQ[0][0];
    for (int i = tid; i < 16 * (HD / 4); i += 256) s[i] = g[i];
  }
  // ---- stage probe-side tables: flat coalesced copies, [f][b] both sides ----
  {
    const float4* a  = (const float4*)wsPr;  float4* da = (float4*)&sPr[0][0];
    const float4* b_ = (const float4*)wsPi;  float4* db = (float4*)&sPi[0][0];
    const float4* c_ = (const float4*)wsW;   float4* dc = (float4*)&sW [0][0];
    const float4* d_ = (const float4*)wsMW;  float4* dd = (float4*)&sMW[0][0];
    for (int i = tid; i < NF * NB / 4; i += 256) {
      da[i] = a[i]; db[i] = b_[i]; dc[i] = c_[i]; dd[i] = d_[i];
    }
  }
  if (tid < NB) sB[tid] = bias[tid];
  __syncthreads();

  // ---- per-row L2 inverse norm ----
  if (tid < 16) {
    float s = 0.0f;
    #pragma unroll 8
    for (int j = 0; j < HD; ++j) { const float v = sQ[tid][j]; s += v * v; }
    sInv[tid] = 1.0f / (sqrtf(s) + EPS);
  }
  __syncthreads();

  // ---- normalized query tables ----
  for (int i = tid; i < NF * 16; i += 256) {
    const int f = i >> 4;
    const int m = i & 15;
    const float inv = sInv[m];
    const float qr = sQ[m][f] * inv;
    const float qi = sQ[m][NF + f] * inv;
    sQri[f][m] = make_float2(qr, qi);
    sQm[f][m]  = sqrtf(qr * qr + qi * qi + EPS);
  }
  __syncthreads();

  const int n    = lane & 15;  // output column within b-tile; also A-frag row M
  const int half = lane >> 4;

  v8f c = {};

  // ---- magnitude term: 16 chained v_wmma_f32_16x16x4_f32 (K=64) ----
  // A (f32 16x4): lanes 0-15 -> M=lane, K=k0,k0+1; lanes 16-31 -> K=k0+2,k0+3.
  // B (f32 4x16): v[0]: K=k0 / K=k0+2 ; v[1]: K=k0+1 / K=k0+3 (N = lane%16).
  for (int f = 0; f < NF; f += 4) {
    v2f av, bv;
    av.x = sQm[f + 2 * half + 0][n];
    av.y = sQm[f + 2 * half + 1][n];
    bv.x = sMW[f + 2 * half + 0][b0 + n];
    bv.y = sMW[f + 2 * half + 1][b0 + n];
    c = __builtin_amdgcn_wmma_f32_16x16x4_f32(
        false, av, false, bv, (short)0, c, false, false);
  }

  // ---- distance term: vectorized LDS reads (4x b128 per f per lane),
  //      accumulating straight into the WMMA C layout (M = r + 8*half). ----
  const float4* qri4 = (const float4*)&sQri[0][0]; // 8 float4 per f-row
  for (int f = 0; f < NF; ++f) {
    const float pr = sPr[f][b0 + n];
    const float pi = sPi[f][b0 + n];
    const float w  = sW [f][b0 + n];
    const int base = f * 8 + 4 * half;    // m = 8*half .. 8*half+7
    const float4 x0 = qri4[base + 0];     // (qr,qi) for m+0, m+1
    const float4 x1 = qri4[base + 1];     // m+2, m+3
    const float4 x2 = qri4[base + 2];     // m+4, m+5
    const float4 x3 = qri4[base + 3];     // m+6, m+7

    float dr, di, t;
#define DIST_ACC(R, QR, QI)                                         \
    dr = pr - (QR); di = pi - (QI);                                 \
    t  = fmaf(dr, dr, fmaf(di, di, EPS));                           \
    c[R] = fmaf(sqrtf(t), w, c[R]);
    DIST_ACC(0, x0.x, x0.y)
    DIST_ACC(1, x0.z, x0.w)
    DIST_ACC(2, x1.x, x1.y)
    DIST_ACC(3, x1.z, x1.w)
    DIST_ACC(4, x2.x, x2.y)
    DIST_ACC(5, x2.z, x2.w)
    DIST_ACC(6, x3.x, x3.y)
    DIST_ACC(7, x3.z, x3.w)
#undef DIST_ACC
  }

  // ---- bias + store ----
  const float bb = sB[b0 + n];
  #pragma unroll
  for (int r = 0; r < 8; ++r) {
    const int m = r + 8 * half;
    out[(size_t)(q0 + m) * NB + b0 + n] = c[r] + bb;
  }
}

// ---------------------------------------------------------------------------
extern "C" void kernel_launch(void* const* d_in, const int* in_sizes, int n_in,
                              void* d_out, int out_size, void* d_ws, size_t ws_size,
                              hipStream_t stream) {
  const float* Q      = (const float*)d_in[0];
  const float* angles = (const float*)d_in[1];
  const float* probes = (const float*)d_in[2];
  const float* w_raw  = (const float*)d_in[3];
  const float* mw     = (const float*)d_in[4];
  const float* bias   = (const float*)d_in[5];
  float* out = (float*)d_out;

  float* wsPr = (float*)d_ws;      // NF*NB each
  float* wsPi = wsPr + NF * NB;
  float* wsW  = wsPi + NF * NB;
  float* wsMW = wsW  + NF * NB;

  prep_kernel<<<NB, NF, 0, stream>>>(probes, angles, w_raw, mw,
                                     wsPr, wsPi, wsW, wsMW);

  qnet_kernel<<<NQ / 16, 256, 0, stream>>>(Q, bias, wsPr, wsPi, wsW, wsMW, out);
}